// LSTM_13125420057288
// MI455X (gfx1250) — compile-verified
//
#include <hip/hip_runtime.h>
#include <math.h>

// ---------------------------------------------------------------------------
// LSTM-cell forward for MI455X (gfx1250, wave32).
// 3 GEMVs [8192 x 16384] @ xc, HBM-bound (1.5 GB weights @ 23.3 TB/s ~ 66us).
// GEMV is done with V_WMMA_F32_16X16X4_F32: broadcast vector in A, weight
// tile in B -> 16 dot-products per wave tile, results in C[0] lanes 0..15.
// K is split 8 ways across waves for memory-level parallelism; partials are
// reduced in a fixed order by a tail kernel (deterministic, no f32 atomics).
// ---------------------------------------------------------------------------

typedef __attribute__((ext_vector_type(2))) float v2f;
typedef __attribute__((ext_vector_type(4))) float f4v;
typedef __attribute__((ext_vector_type(8))) float v8f;

#define IN_F   8192
#define OUT_F  8192
#define CAT    16384
#define KSPLIT 8
#define KSLICE (CAT / KSPLIT)   // 2048, entirely inside x-half or pred-half
#define TILES  (OUT_F / 16)     // 512 row tiles per gate
#define WPB    8                // waves per 256-thread block

// One wave: 16 output rows of one gate, one K-slice of 2048.
// Per 8-k chunk: lane half h in {0,1}; lane loads W[row][kb+4h .. kb+4h+3]
// (b128, contiguous) and xc[kb+4h .. kb+4h+3] (broadcast, cache-hot).
// WMMA #1 consumes k-values {kb,kb+1, kb+4,kb+5} (slots K0,K1 | K2,K3),
// WMMA #2 consumes {kb+2,kb+3, kb+6,kb+7}.  A and B use the same permutation,
// so the accumulated dot product is exact.
__global__ __launch_bounds__(256) void lstm_gemv_wmma(
    const float* __restrict__ x, const float* __restrict__ pred,
    const float* __restrict__ Wf, const float* __restrict__ Wi,
    const float* __restrict__ Wo, float* __restrict__ part)
{
    const int lane = threadIdx.x & 31;
    const int gw   = blockIdx.x * WPB + (threadIdx.x >> 5);   // 0..12287
    const int ks   = gw & (KSPLIT - 1);
    const int tile = (gw >> 3) & (TILES - 1);
    const int gate = gw >> 12;                                // 0..2

    const float* __restrict__ W = (gate == 0) ? Wf : (gate == 1) ? Wi : Wo;

    const int h = lane >> 4;       // half-wave id
    const int n = lane & 15;       // row-within-tile / output column

    const float* __restrict__ wrow =
        W + (size_t)(tile * 16 + n) * CAT + (size_t)ks * KSLICE + 4 * h;
    const float* __restrict__ vbase =
        ((ks < KSPLIT / 2) ? (x + (size_t)ks * KSLICE)
                           : (pred + (size_t)(ks - KSPLIT / 2) * KSLICE)) + 4 * h;

    v8f acc0 = {};
    v8f acc1 = {};

    for (int c = 0; c < KSLICE; c += 64) {
#pragma unroll
        for (int j = 0; j < 8; ++j) {
            const int kb = c + j * 8;
            // Streaming weights: non-temporal b128 load.
            f4v w  = __builtin_nontemporal_load((const f4v*)(wrow + kb));
            // Shared vector: temporal load (hot in WGP$/L2).
            f4v xv = *(const f4v*)(vbase + kb);

            v2f a0 = { xv.x, xv.y };
            v2f b0 = { w.x,  w.y  };
            v2f a1 = { xv.z, xv.w };
            v2f b1 = { w.z,  w.w  };

            acc0 = __builtin_amdgcn_wmma_f32_16x16x4_f32(
                false, a0, false, b0, (short)0, acc0, false, false);
            acc1 = __builtin_amdgcn_wmma_f32_16x16x4_f32(
                false, a1, false, b1, (short)0, acc1, false, false);
        }
    }

    // D[m][n] identical over m (A rows are identical broadcasts of xc),
    // so C VGPR0 lanes 0..15 hold the 16 partial dot products.
    if (lane < 16) {
        part[(size_t)ks * (3 * OUT_F) + (size_t)gate * OUT_F + tile * 16 + n] =
            acc0[0] + acc1[0];
    }
}

__global__ __launch_bounds__(256) void lstm_tail(
    const float* __restrict__ part, const float* __restrict__ bf,
    const float* __restrict__ bi, const float* __restrict__ bo,
    const float* __restrict__ hs, float* __restrict__ out)
{
    const int r = blockIdx.x * 256 + threadIdx.x;   // 0..8191
    float lf = bf[r], li = bi[r], lo = bo[r];
#pragma unroll
    for (int ks = 0; ks < KSPLIT; ++ks) {           // fixed order: deterministic
        const float* p = part + (size_t)ks * (3 * OUT_F);
        lf += p[r];
        li += p[OUT_F + r];
        lo += p[2 * OUT_F + r];
    }
    const float fg = 1.0f / (1.0f + expf(-lf));                 // sigmoid
    const float ig = tanhf(li) * (1.0f / (1.0f + expf(-li)));   // tanh*sigmoid
    const float og = 1.0f / (1.0f + expf(-lo));
    const float hh = fg * hs[r] + ig;
    out[r] = og * fmaxf(hh, 0.0f);
}

extern "C" void kernel_launch(void* const* d_in, const int* in_sizes, int n_in,
                              void* d_out, int out_size, void* d_ws, size_t ws_size,
                              hipStream_t stream) {
    // setup_inputs() order:
    // 0:x 1:prediction 2:hidden_state 3:W_f 4:b_f 5:W_i 6:b_i 7:W_bi 8:b_bi 9:W_o 10:b_o
    const float* x    = (const float*)d_in[0];
    const float* pred = (const float*)d_in[1];
    const float* hs   = (const float*)d_in[2];
    const float* Wf   = (const float*)d_in[3];
    const float* bf   = (const float*)d_in[4];
    const float* Wi   = (const float*)d_in[5];
    const float* bi   = (const float*)d_in[6];
    // d_in[7], d_in[8] (W_bi, b_bi) are dead code in the reference.
    const float* Wo   = (const float*)d_in[9];
    const float* bo   = (const float*)d_in[10];

    float* part = (float*)d_ws;   // KSPLIT * 3 * OUT_F floats = 768 KB

    const int total_waves = 3 * TILES * KSPLIT;   // 12288
    const int blocks = total_waves / WPB;         // 1536

    lstm_gemv_wmma<<<blocks, 256, 0, stream>>>(x, pred, Wf, Wi, Wo, part);
    lstm_tail<<<OUT_F / 256, 256, 0, stream>>>(part, bf, bi, bo, hs, (float*)d_out);
}